// MultiTaskLossPro_4844723110172
// MI455X (gfx1250) — compile-verified
//
#include <hip/hip_runtime.h>
#include <hip/hip_bf16.h>
#include <math.h>
#include <stdint.h>

#define NPT 2000
#define DIM 352
#define KNN 64
#define NUM_REGION 6
#define TEMP_INV (1.0f / 0.07f)
#define GAMMA 0.5f
#define SMOOTH 0.1f

// macro tile: 64 rows x 128 cols per 256-thread block, K staged in chunks of 32
#define MT_R 64
#define MT_C 128
#define KC 32
#define PAN_R ((NPT + MT_R - 1) / MT_R)   // 32 (covers 2048)
#define PAN_C ((NPT + MT_C - 1) / MT_C)   // 16 (covers 2048)

typedef __attribute__((ext_vector_type(2))) float v2f;
typedef __attribute__((ext_vector_type(8))) float v8f;
typedef unsigned int u32x4 __attribute__((ext_vector_type(4)));
typedef int i32x4 __attribute__((ext_vector_type(4)));
typedef int i32x8 __attribute__((ext_vector_type(8)));

__constant__ float c_class_w[6] = {0.3f, 1.5f, 1.0f, 1.0f, 1.0f, 1.0f};

__device__ __forceinline__ float block_reduce_sum(float v, float* buf) {
    int t = threadIdx.x;
    buf[t] = v;
    __syncthreads();
    for (int s = blockDim.x >> 1; s > 0; s >>= 1) {
        if (t < s) buf[t] += buf[t + s];
        __syncthreads();
    }
    float r = buf[0];
    __syncthreads();
    return r;
}

// ---------------------------------------------------------------------------
// TDM: issue a 2D tile load (tile_rows x tile_cols f32, row stride
// row_stride_elems) from global into LDS at lds_addr. Tile-relative tensor
// dims {rem_cols, rem_rows} make the TDM zero-fill out-of-bounds rows.
// D# layout per CDNA5 ISA sec 8.3-8.5.
// ---------------------------------------------------------------------------
#if __has_builtin(__builtin_amdgcn_tensor_load_to_lds)
#define HAVE_TDM 1
__device__ __forceinline__ void tdm_load_tile_2d(
        unsigned lds_addr, const float* gptr,
        unsigned rem_cols, unsigned rem_rows,
        unsigned tile_cols, unsigned tile_rows, unsigned row_stride_elems) {
    unsigned long long ga = (unsigned long long)(uintptr_t)gptr;
    u32x4 g0;
    g0[0] = 1u;                                        // count=1 (valid D#)
    g0[1] = lds_addr;                                  // lds_addr [63:32]
    g0[2] = (unsigned)(ga & 0xFFFFFFFFull);            // global_addr lo
    g0[3] = (unsigned)((ga >> 32) & 0x1FFFFFFull)      // global_addr [56:32]
            | (2u << 30);                              // type=2 ("image")
    i32x8 g1;
    g1[0] = (int)(2u << 16);                           // data_size=2 -> 4B
    g1[1] = (int)((rem_cols & 0xFFFFu) << 16);         // tensor_dim0 lo16
    g1[2] = (int)((rem_cols >> 16) | ((rem_rows & 0xFFFFu) << 16));
    g1[3] = (int)((rem_rows >> 16) | (tile_cols << 16));   // tile_dim0
    g1[4] = (int)tile_rows;                            // tile_dim1 (tile_dim2=0)
    g1[5] = (int)row_stride_elems;                     // tensor_dim0_stride lo32
    g1[6] = 0;
    g1[7] = 0;
    i32x4 z4 = {0, 0, 0, 0};
#if defined(__clang_major__) && (__clang_major__ >= 23)
    i32x8 z8 = {0, 0, 0, 0, 0, 0, 0, 0};
    __builtin_amdgcn_tensor_load_to_lds(g0, g1, z4, z4, z8, 0);
#else
    __builtin_amdgcn_tensor_load_to_lds(g0, g1, z4, z4, 0);
#endif
}
#else
#define HAVE_TDM 0
#endif

// ---------------------------------------------------------------------------
// Kernel 1: G = E * E^T via V_WMMA_F32_16X16X4_F32, LDS-staged macro tiles.
// Block (8 waves) computes 64x128 of G; TDM stages A(64xKC) and B(128xKC)
// panels into LDS each K-chunk; each wave computes a 16x64 strip (4 accums).
// ---------------------------------------------------------------------------
__global__ __launch_bounds__(256) void gram_wmma_kernel(
        const float* __restrict__ E, float* __restrict__ G) {
    __shared__ float As[MT_R][KC];    // 8 KB
    __shared__ float Bs[MT_C][KC];    // 16 KB

    const int panel_r = blockIdx.x / PAN_C;
    const int panel_c = blockIdx.x % PAN_C;
    const int w    = threadIdx.x >> 5;      // wave 0..7
    const int lane = threadIdx.x & 31;
    const int hv   = lane >> 4;             // K-pair select per WMMA layout
    const int m    = lane & 15;
    const int tr   = w >> 1;                // tile row 0..3
    const int ch   = w & 1;                 // column half 0..1

    const unsigned rem_rows_a = (unsigned)(NPT - panel_r * MT_R);
    const unsigned rem_rows_b = (unsigned)(NPT - panel_c * MT_C);

    v8f acc[4];
#pragma unroll
    for (int t = 0; t < 4; ++t)
        acc[t] = (v8f){0.f, 0.f, 0.f, 0.f, 0.f, 0.f, 0.f, 0.f};

    for (int k0 = 0; k0 < DIM; k0 += KC) {
#if HAVE_TDM
        if (threadIdx.x < 32) {             // wave 0 issues both DMAs
            tdm_load_tile_2d((unsigned)(uintptr_t)&As[0][0],
                             E + (size_t)(panel_r * MT_R) * DIM + k0,
                             (unsigned)(DIM - k0), rem_rows_a, KC, MT_R, DIM);
            tdm_load_tile_2d((unsigned)(uintptr_t)&Bs[0][0],
                             E + (size_t)(panel_c * MT_C) * DIM + k0,
                             (unsigned)(DIM - k0), rem_rows_b, KC, MT_C, DIM);
            __builtin_amdgcn_s_wait_tensorcnt(0);
        }
        __syncthreads();
#else
        // synchronous fallback staging (float4 per thread, zero-fill OOB rows)
        for (int idx = threadIdx.x; idx < MT_R * (KC / 4); idx += 256) {
            int r = idx >> 3, c4 = (idx & 7) * 4;
            int grow = panel_r * MT_R + r;
            float4 v = {0.f, 0.f, 0.f, 0.f};
            if (grow < NPT) v = *(const float4*)&E[(size_t)grow * DIM + k0 + c4];
            *(float4*)&As[r][c4] = v;
        }
        for (int idx = threadIdx.x; idx < MT_C * (KC / 4); idx += 256) {
            int r = idx >> 3, c4 = (idx & 7) * 4;
            int grow = panel_c * MT_C + r;
            float4 v = {0.f, 0.f, 0.f, 0.f};
            if (grow < NPT) v = *(const float4*)&E[(size_t)grow * DIM + k0 + c4];
            *(float4*)&Bs[r][c4] = v;
        }
        __syncthreads();
#endif
#pragma unroll
        for (int kk = 0; kk < KC; kk += 4) {
            int kc = kk + 2 * hv;
            v2f a = *(const v2f*)&As[tr * 16 + m][kc];
#pragma unroll
            for (int t = 0; t < 4; ++t) {
                v2f b = *(const v2f*)&Bs[ch * 64 + t * 16 + m][kc];
                acc[t] = __builtin_amdgcn_wmma_f32_16x16x4_f32(
                        false, a, false, b, (short)0, acc[t], false, false);
            }
        }
        __syncthreads();                    // protect LDS before next stage
    }

    // store 16x64 strip (per-lane guards AFTER all WMMAs: EXEC all-1s above)
    const int gr0 = panel_r * MT_R + tr * 16 + hv * 8;
#pragma unroll
    for (int t = 0; t < 4; ++t) {
        int gc = panel_c * MT_C + ch * 64 + t * 16 + m;
#pragma unroll
        for (int r = 0; r < 8; ++r) {
            int gr = gr0 + r;
            if (gr < NPT && gc < NPT)
                G[(size_t)gr * NPT + gc] = acc[t][r];
        }
    }
}

// ---------------------------------------------------------------------------
// Kernel 2: per-row 64-NN selection in 2D coord space (matches top_k order:
// ascending distance, lowest index on ties; slot 0 = self, skipped).
// ---------------------------------------------------------------------------
__global__ __launch_bounds__(256) void knn_kernel(
        const float* __restrict__ coords, int* __restrict__ knn) {
    __shared__ float sd[NPT];
    __shared__ float rv[256];
    __shared__ int   ri[256];
    int i = blockIdx.x;
    float cx = coords[2 * i], cy = coords[2 * i + 1];
    for (int j = threadIdx.x; j < NPT; j += 256) {
        float dx = cx - coords[2 * j], dy = cy - coords[2 * j + 1];
        sd[j] = sqrtf(dx * dx + dy * dy);
    }
    __syncthreads();
    for (int t = 0; t <= KNN; ++t) {
        float best = __builtin_inff();
        int   bi   = NPT;
        for (int j = threadIdx.x; j < NPT; j += 256) {
            float v = sd[j];
            if (v < best) { best = v; bi = j; }
        }
        rv[threadIdx.x] = best;
        ri[threadIdx.x] = bi;
        __syncthreads();
        for (int s = 128; s > 0; s >>= 1) {
            if (threadIdx.x < s) {
                float ov = rv[threadIdx.x + s];
                int   oi = ri[threadIdx.x + s];
                if (ov < rv[threadIdx.x] ||
                    (ov == rv[threadIdx.x] && oi < ri[threadIdx.x])) {
                    rv[threadIdx.x] = ov;
                    ri[threadIdx.x] = oi;
                }
            }
            __syncthreads();
        }
        if (threadIdx.x == 0) {
            int sel = ri[0];
            if (t > 0) knn[i * KNN + (t - 1)] = sel;
            sd[sel] = __builtin_inff();
        }
        __syncthreads();
    }
}

// ---------------------------------------------------------------------------
// Kernel 3: per-row pass over G -> Pearson raw moments + contrastive row loss.
// rs[i*8 + {0..6}] = {sum_ed, sum_cd, sum_ed*cd, sum_ed2, sum_cd2, loss*w, w}
// ---------------------------------------------------------------------------
__global__ __launch_bounds__(256) void rowstats_kernel(
        const float* __restrict__ G, const float* __restrict__ coords,
        const int* __restrict__ rlab, float* __restrict__ rs) {
    __shared__ float buf[256];
    int i = blockIdx.x;
    float gii = G[(size_t)i * (NPT + 1)];
    float ni  = sqrtf(gii);
    float cx = coords[2 * i], cy = coords[2 * i + 1];
    int  li = rlab[i];
    bool vi = (li < NUM_REGION);

    float s_ed = 0.f, s_cd = 0.f, s_xy = 0.f, s_x2 = 0.f, s_y2 = 0.f;
    float pos = 0.f, all = 0.f;
    for (int j = threadIdx.x; j < NPT; j += 256) {
        float gij = G[(size_t)i * NPT + j];
        float gjj = G[(size_t)j * (NPT + 1)];
        float ed  = sqrtf(fmaxf(gii + gjj - 2.f * gij, 0.f));
        float dx = cx - coords[2 * j], dy = cy - coords[2 * j + 1];
        float cd = sqrtf(dx * dx + dy * dy);
        s_ed += ed;  s_cd += cd;  s_xy += ed * cd;
        s_x2 += ed * ed;  s_y2 += cd * cd;
        if (j != i) {
            float nj  = sqrtf(gjj);
            float sim = gij / (ni * nj) * TEMP_INV;
            float es  = expf(sim);
            all += es;
            if (vi && rlab[j] == li) pos += es;
        }
    }
    s_ed = block_reduce_sum(s_ed, buf);
    s_cd = block_reduce_sum(s_cd, buf);
    s_xy = block_reduce_sum(s_xy, buf);
    s_x2 = block_reduce_sum(s_x2, buf);
    s_y2 = block_reduce_sum(s_y2, buf);
    pos  = block_reduce_sum(pos, buf);
    all  = block_reduce_sum(all, buf);
    if (threadIdx.x == 0) {
        float has_pos = (pos > 0.f) ? 1.f : 0.f;
        int safe = min(max(li, 0), NUM_REGION - 1);
        float w = c_class_w[safe] * has_pos;
        float loss = -logf(pos / (all + 1e-8f) + 1e-8f);
        rs[i * 8 + 0] = s_ed;  rs[i * 8 + 1] = s_cd;  rs[i * 8 + 2] = s_xy;
        rs[i * 8 + 3] = s_x2;  rs[i * 8 + 4] = s_y2;
        rs[i * 8 + 5] = loss * w;  rs[i * 8 + 6] = w;
    }
}

// ---------------------------------------------------------------------------
// Kernel 4: local kNN distance loss; pred_d gathered from G via knn indices.
// ---------------------------------------------------------------------------
__global__ __launch_bounds__(256) void local_kernel(
        const float* __restrict__ G, const float* __restrict__ coords,
        const int* __restrict__ knn, float* __restrict__ lp) {
    __shared__ float buf[256];
    int p = blockIdx.x * 256 + threadIdx.x;   // exactly NPT*KNN threads
    int i = p >> 6;
    int j = knn[p];
    float gii = G[(size_t)i * (NPT + 1)];
    float gjj = G[(size_t)j * (NPT + 1)];
    float gij = G[(size_t)i * NPT + j];
    float pred_d = sqrtf(fmaxf(gii + gjj - 2.f * gij, 0.f));
    float dx = coords[2 * i] - coords[2 * j];
    float dy = coords[2 * i + 1] - coords[2 * j + 1];
    float true_d = sqrtf(dx * dx + dy * dy);
    float diff = pred_d - true_d;
    float v = diff * diff * expf(-GAMMA * true_d);
    v = block_reduce_sum(v, buf);
    if (threadIdx.x == 0) lp[blockIdx.x] = v;
}

// ---------------------------------------------------------------------------
// Kernel 5: center loss per row.
// ---------------------------------------------------------------------------
__global__ __launch_bounds__(256) void center_kernel(
        const float* __restrict__ E, const float* __restrict__ centers,
        const int* __restrict__ rlab, float* __restrict__ cs) {
    __shared__ float buf[256];
    int i = blockIdx.x;
    int l = rlab[i];
    float valid = (l < NUM_REGION) ? 1.f : 0.f;
    int safe = min(max(l, 0), NUM_REGION - 1);
    float s = 0.f;
    for (int d = threadIdx.x; d < DIM; d += 256) {
        float diff = E[(size_t)i * DIM + d] - centers[(size_t)safe * DIM + d];
        s += diff * diff;
    }
    s = block_reduce_sum(s, buf);
    if (threadIdx.x == 0) {
        cs[i * 2 + 0] = s * valid;
        cs[i * 2 + 1] = valid;
    }
}

// ---------------------------------------------------------------------------
// Kernel 6: per-row small losses: recon MSE terms, direction CE, region CE.
// ---------------------------------------------------------------------------
__global__ __launch_bounds__(256) void smallce_kernel(
        const float* __restrict__ coords, const float* __restrict__ pred,
        const float* __restrict__ dlog, const int* __restrict__ dlab,
        const float* __restrict__ rlog, const int* __restrict__ rlab,
        float* __restrict__ sc) {
    int i = blockIdx.x * 256 + threadIdx.x;
    if (i >= NPT) return;
    float r0 = pred[2 * i] - coords[2 * i];
    float r1 = pred[2 * i + 1] - coords[2 * i + 1];
    float rec = r0 * r0 + r1 * r1;

    float mx = -__builtin_inff(), sx = 0.f, se = 0.f;
    for (int k = 0; k < 8; ++k) mx = fmaxf(mx, dlog[i * 8 + k]);
    for (int k = 0; k < 8; ++k) {
        float x = dlog[i * 8 + k];
        sx += x;
        se += expf(x - mx);
    }
    float lse = mx + logf(se);
    float nll = lse - dlog[i * 8 + dlab[i]];
    float smo = lse - sx * (1.f / 8.f);
    float dir_ce = (1.f - SMOOTH) * nll + SMOOTH * smo;

    int l = rlab[i];
    float valid = (l < NUM_REGION) ? 1.f : 0.f;
    int safe = min(max(l, 0), NUM_REGION - 1);
    mx = -__builtin_inff(); sx = 0.f; se = 0.f;
    for (int k = 0; k < 6; ++k) mx = fmaxf(mx, rlog[i * 6 + k]);
    for (int k = 0; k < 6; ++k) {
        float x = rlog[i * 6 + k];
        sx += x;
        se += expf(x - mx);
    }
    lse = mx + logf(se);
    nll = lse - rlog[i * 6 + safe];
    smo = lse - sx * (1.f / 6.f);
    float reg_ce = (1.f - SMOOTH) * nll + SMOOTH * smo;

    sc[i * 4 + 0] = rec;
    sc[i * 4 + 1] = dir_ce;
    sc[i * 4 + 2] = reg_ce * valid;
    sc[i * 4 + 3] = valid;
}

// ---------------------------------------------------------------------------
// Kernel 7: deterministic final reduction + weighted total.
// ---------------------------------------------------------------------------
__global__ __launch_bounds__(256) void finalize_kernel(
        const float* __restrict__ rs, const float* __restrict__ cs,
        const float* __restrict__ sc, const float* __restrict__ lp,
        float* __restrict__ out) {
    __shared__ float buf[256];
    __shared__ float q[16];
    int t = threadIdx.x;
    for (int k = 0; k < 7; ++k) {
        float s = 0.f;
        for (int i = t; i < NPT; i += 256) s += rs[i * 8 + k];
        s = block_reduce_sum(s, buf);
        if (t == 0) q[k] = s;
        __syncthreads();
    }
    for (int k = 0; k < 2; ++k) {
        float s = 0.f;
        for (int i = t; i < NPT; i += 256) s += cs[i * 2 + k];
        s = block_reduce_sum(s, buf);
        if (t == 0) q[7 + k] = s;
        __syncthreads();
    }
    for (int k = 0; k < 4; ++k) {
        float s = 0.f;
        for (int i = t; i < NPT; i += 256) s += sc[i * 4 + k];
        s = block_reduce_sum(s, buf);
        if (t == 0) q[9 + k] = s;
        __syncthreads();
    }
    {
        float s = 0.f;
        for (int i = t; i < (NPT * KNN) / 256; i += 256) s += lp[i];
        s = block_reduce_sum(s, buf);
        if (t == 0) q[13] = s;
        __syncthreads();
    }
    if (t == 0) {
        const float invNN = 1.f / ((float)NPT * (float)NPT);
        float mx = q[0] * invNN, my = q[1] * invNN;
        float cov = q[2] * invNN - mx * my;
        float sx = sqrtf(q[3] * invNN - mx * mx + 1e-8f);
        float sy = sqrtf(q[4] * invNN - my * my + 1e-8f);
        float pearson = cov / (sx * sy + 1e-8f);
        float l_dist = (1.f - pearson) +
                       0.5f * q[13] / ((float)NPT * (float)KNN);
        float l_region = q[5] / fmaxf(q[6], 1e-8f);
        float l_center = q[7] / fmaxf(q[8], 1.f);
        float l_recon = q[9] / ((float)NPT * 2.f);
        float l_dir = q[10] / (float)NPT;
        float l_region_clf = q[11] / fmaxf(q[12], 1.f);
        out[0] = 3.0f * l_dist + 1.0f * l_recon + 1.0f * l_dir +
                 0.3f * l_region + 1.0f * l_region_clf + 0.5f * l_center;
    }
}

// ---------------------------------------------------------------------------
extern "C" void kernel_launch(void* const* d_in, const int* in_sizes, int n_in,
                              void* d_out, int out_size, void* d_ws, size_t ws_size,
                              hipStream_t stream) {
    const float* emb     = (const float*)d_in[0];
    const float* coords  = (const float*)d_in[1];
    const float* pred    = (const float*)d_in[2];
    const float* dlog    = (const float*)d_in[3];
    const int*   dlab    = (const int*)d_in[4];
    const float* rlog    = (const float*)d_in[5];
    const int*   rlab    = (const int*)d_in[6];
    const float* centers = (const float*)d_in[7];

    // workspace layout (~16.7 MB total)
    char*  ws  = (char*)d_ws;
    float* G   = (float*)ws;                                     // NPT*NPT f32
    int*   knn = (int*)(ws + (size_t)NPT * NPT * 4);             // NPT*KNN i32
    float* rs  = (float*)(ws + (size_t)NPT * NPT * 4 +
                          (size_t)NPT * KNN * 4);                // NPT*8
    float* cs  = rs + NPT * 8;                                   // NPT*2
    float* sc  = cs + NPT * 2;                                   // NPT*4
    float* lp  = sc + NPT * 4;                                   // 500

    gram_wmma_kernel<<<PAN_R * PAN_C, 256, 0, stream>>>(emb, G);
    knn_kernel<<<NPT, 256, 0, stream>>>(coords, knn);
    rowstats_kernel<<<NPT, 256, 0, stream>>>(G, coords, rlab, rs);
    local_kernel<<<(NPT * KNN) / 256, 256, 0, stream>>>(G, coords, knn, lp);
    center_kernel<<<NPT, 256, 0, stream>>>(emb, centers, rlab, cs);
    smallce_kernel<<<(NPT + 255) / 256, 256, 0, stream>>>(
        coords, pred, dlog, dlab, rlog, rlab, sc);
    finalize_kernel<<<1, 256, 0, stream>>>(rs, cs, sc, lp, (float*)d_out);
}